// AttentionDecoderBatch_56358560858502
// MI455X (gfx1250) — compile-verified
//
#include <hip/hip_runtime.h>
#include <hip/hip_bf16.h>
#include <math.h>

#define Hdim 512
#define Bdim 4096
#define Mdim 32

typedef __attribute__((ext_vector_type(2))) float v2f;
typedef __attribute__((ext_vector_type(8))) float v8f;

// ---------------------------------------------------------------------------
// Phase 1: Qin[b,:] = h_static[cur[b],:] + h_dynamic[cur[b],:]
// ---------------------------------------------------------------------------
__global__ void gather_sum_kernel(const float* __restrict__ hs,
                                  const float* __restrict__ hd,
                                  const int* __restrict__ cur,
                                  float* __restrict__ Qin) {
    int b = blockIdx.x;
    size_t n = (size_t)cur[b];
    const float* rs = hs + n * Hdim;
    const float* rd = hd + n * Hdim;
    float* q = Qin + (size_t)b * Hdim;
    for (int h = threadIdx.x; h < Hdim; h += blockDim.x)
        q[h] = rs[h] + rd[h];
}

// ---------------------------------------------------------------------------
// Phase 2: fp32 WMMA GEMM  C[M x N] = A[M x K] @ B (optionally B accessed
// transposed: Bmat[k,n] = Bm[n*ldb + k]).  One wave computes one 16x16 tile,
// using V_WMMA_F32_16X16X4_F32 (exact fp32, ISA 7.12 layouts).
// ---------------------------------------------------------------------------
template <bool TRANSB>
__global__ void wmma_gemm_f32(const float* __restrict__ A, int lda,
                              const float* __restrict__ Bm, int ldb,
                              float* __restrict__ C, int ldc,
                              int tilesM, int tilesN, int K) {
    int wave = blockIdx.x * (blockDim.x >> 5) + (threadIdx.x >> 5);
    if (wave >= tilesM * tilesN) return;
    int tm = wave / tilesN;
    int tn = wave % tilesN;

    int lane = threadIdx.x & 31;
    int half = lane >> 4;      // 0: K=0,1  1: K=2,3
    int r    = lane & 15;      // A row / B col within tile

    size_t arow = (size_t)(tm * 16 + r);
    size_t bcol = (size_t)(tn * 16 + r);

    v8f acc = {};
#pragma unroll 4
    for (int k = 0; k < K; k += 4) {
        int ka = k + 2 * half;
        v2f a, b;
        a.x = A[arow * lda + ka];
        a.y = A[arow * lda + ka + 1];
        if (TRANSB) {
            b.x = Bm[bcol * ldb + ka];
            b.y = Bm[bcol * ldb + ka + 1];
        } else {
            b.x = Bm[(size_t)ka * ldb + bcol];
            b.y = Bm[(size_t)(ka + 1) * ldb + bcol];
        }
        acc = __builtin_amdgcn_wmma_f32_16x16x4_f32(
            /*neg_a=*/false, a, /*neg_b=*/false, b,
            /*c_mod=*/(short)0, acc, /*reuse_a=*/false, /*reuse_b=*/false);
    }
#pragma unroll
    for (int v = 0; v < 8; ++v) {
        size_t row = (size_t)(tm * 16 + v + 8 * half);
        C[row * ldc + tn * 16 + r] = acc[v];
    }
}

// ---------------------------------------------------------------------------
// Threefry2x32 with key (0,1)  == jax.random.key(1); counts [i, 65536+i].
// ---------------------------------------------------------------------------
__device__ __forceinline__ unsigned rotl32(unsigned x, int d) {
    return (x << d) | (x >> (32 - d));
}

__device__ __forceinline__ unsigned threefry_u32(unsigned i, bool hi) {
    unsigned ks[3] = {0u, 1u, 0u ^ 1u ^ 0x1BD11BDAu};
    unsigned x0 = i + ks[0];
    unsigned x1 = (65536u + i) + ks[1];
    const int rotA[4] = {13, 15, 26, 6};
    const int rotB[4] = {17, 29, 16, 24};
#pragma unroll
    for (int blk = 0; blk < 5; ++blk) {
        const int* rot = (blk & 1) ? rotB : rotA;
#pragma unroll
        for (int j = 0; j < 4; ++j) {
            x0 += x1;
            x1 = rotl32(x1, rot[j]);
            x1 ^= x0;
        }
        x0 += ks[(blk + 1) % 3];
        x1 += ks[(blk + 2) % 3] + (unsigned)(blk + 1);
    }
    return hi ? x1 : x0;
}

// ---------------------------------------------------------------------------
// Phase 3: per batch element b (one 256-thread block = 8 waves):
//   compat[m] = (h_s[n]·QS[b] + h_d[n]·QD[b]) / sqrt(H)
//   Gumbel argmax + log-softmax + empty-neighborhood handling.
// ---------------------------------------------------------------------------
__global__ void score_sample_kernel(const float* __restrict__ hs,
                                    const float* __restrict__ hd,
                                    const float* __restrict__ QS,
                                    const float* __restrict__ QD,
                                    const int* __restrict__ neigh,
                                    const int* __restrict__ lengths,
                                    const int* __restrict__ cur,
                                    float* __restrict__ out) {
    __shared__ float s_qs[Hdim];
    __shared__ float s_qd[Hdim];
    __shared__ float s_comp[Mdim];

    int b = blockIdx.x;
    for (int t = threadIdx.x; t < Hdim; t += blockDim.x) {
        s_qs[t] = QS[(size_t)b * Hdim + t];
        s_qd[t] = QD[(size_t)b * Hdim + t];
    }
    __syncthreads();

    int wave = threadIdx.x >> 5;
    int lane = threadIdx.x & 31;
    const float inv_sqrt_h = 0.044194173824159216f; // 1/sqrt(512)

    for (int m = wave; m < Mdim; m += 8) {
        size_t n = (size_t)neigh[(size_t)b * Mdim + m];
        const float* rs = hs + n * Hdim;
        const float* rd = hd + n * Hdim;
        float acc = 0.0f;
#pragma unroll 4
        for (int h = lane; h < Hdim; h += 32)
            acc = fmaf(rs[h], s_qs[h], fmaf(rd[h], s_qd[h], acc));
#pragma unroll
        for (int off = 16; off; off >>= 1)
            acc += __shfl_xor(acc, off, 32);
        if (lane == 0) s_comp[m] = acc * inv_sqrt_h;
    }
    __syncthreads();

    if (threadIdx.x < 32) {
        int m = threadIdx.x;
        int L = lengths[b];
        float logit = (m < L) ? s_comp[m] : -1.0e9f;

        // u ~ jax.random.uniform(key(1), (B,M), f32, 1e-8, 1.0)
        unsigned j = (unsigned)b * 32u + (unsigned)m;
        bool hi = (j >= 65536u);
        unsigned bits = threefry_u32(hi ? (j - 65536u) : j, hi);
        float f = __uint_as_float((bits >> 9) | 0x3F800000u); // [1,2)
        const float minv = 1.0e-8f;
        float u = fmaxf(minv, f * (1.0f - minv) + (minv - (1.0f - minv)));
        float g = -logf(-logf(u));

        // argmax(logits + g), first-occurrence tie-break
        float bv = logit + g;
        int bi = m;
#pragma unroll
        for (int off = 16; off; off >>= 1) {
            float ov = __shfl_xor(bv, off, 32);
            int   oi = __shfl_xor(bi, off, 32);
            if (ov > bv || (ov == bv && oi < bi)) { bv = ov; bi = oi; }
        }
        // log-softmax of logits
        float mx = logit;
#pragma unroll
        for (int off = 16; off; off >>= 1)
            mx = fmaxf(mx, __shfl_xor(mx, off, 32));
        float e = expf(logit - mx);
#pragma unroll
        for (int off = 16; off; off >>= 1)
            e += __shfl_xor(e, off, 32);

        float sel = __shfl(logit, bi, 32);
        if (m == 0) {
            float lp = sel - mx - logf(e);
            int act = (L > 0) ? neigh[(size_t)b * Mdim + bi] : cur[b];
            out[b]        = (float)act;
            out[Bdim + b] = (L > 0) ? lp : 0.0f;
        }
    }
}

// ---------------------------------------------------------------------------
extern "C" void kernel_launch(void* const* d_in, const int* in_sizes, int n_in,
                              void* d_out, int out_size, void* d_ws, size_t ws_size,
                              hipStream_t stream) {
    const float* h_s  = (const float*)d_in[0];
    const float* h_d  = (const float*)d_in[1];
    const float* Ws   = (const float*)d_in[2]; // 512 x 1536
    const float* Wd   = (const float*)d_in[3]; // 512 x 1536
    const float* Wq   = (const float*)d_in[4]; // 512 x 512
    // d_in[5] = batch (unused by reference)
    const int*   cur   = (const int*)d_in[6];
    const int*   neigh = (const int*)d_in[7];
    const int*   len   = (const int*)d_in[8];
    float* out = (float*)d_out;

    const size_t mat = (size_t)Bdim * Hdim; // 2M floats
    float* Qin = (float*)d_ws;
    float* Q   = Qin + mat;
    float* QS  = Q   + mat;
    float* QD  = QS  + mat;

    gather_sum_kernel<<<Bdim, 256, 0, stream>>>(h_s, h_d, cur, Qin);

    const int tilesM = Bdim / 16;  // 256
    const int tilesN = Hdim / 16;  // 32
    const int waves  = tilesM * tilesN;          // 8192 waves
    const int blocks = waves / 8;                // 8 waves per 256-thr block

    // Q = Qin @ Wq          (Wq row-major, no transpose)
    wmma_gemm_f32<false><<<blocks, 256, 0, stream>>>(Qin, Hdim, Wq, Hdim,
                                                     Q, Hdim, tilesM, tilesN, Hdim);
    // QS = Q @ Ws_k^T       (Ws_k = Ws_kvl[:, :512], ldb = 1536, transposed)
    wmma_gemm_f32<true><<<blocks, 256, 0, stream>>>(Q, Hdim, Ws, 3 * Hdim,
                                                    QS, Hdim, tilesM, tilesN, Hdim);
    // QD = Q @ Wd_k^T
    wmma_gemm_f32<true><<<blocks, 256, 0, stream>>>(Q, Hdim, Wd, 3 * Hdim,
                                                    QD, Hdim, tilesM, tilesN, Hdim);

    score_sample_kernel<<<Bdim, 256, 0, stream>>>(h_s, h_d, QS, QD,
                                                  neigh, len, cur, out);
}